// GCNLayer_22376779612463
// MI455X (gfx1250) — compile-verified
//
#include <hip/hip_runtime.h>
#include <hip/hip_bf16.h>

typedef __attribute__((ext_vector_type(2))) float v2f;
typedef __attribute__((ext_vector_type(8))) float v8f;

// ---------------------------------------------------------------------------
// Degree kernels: deg[n] starts at 1.0 (self-loop weight), edges atomically
// accumulate, then deg -> deg^{-1/2} in place.
// ---------------------------------------------------------------------------
__global__ void deg_init_kernel(float* __restrict__ deg, int n) {
    int i = blockIdx.x * blockDim.x + threadIdx.x;
    if (i < n) deg[i] = 1.0f;
}

__global__ void deg_accum_kernel(const int* __restrict__ dst,
                                 const float* __restrict__ w,
                                 float* __restrict__ deg, int e) {
    int i = blockIdx.x * blockDim.x + threadIdx.x;
    if (i < e) atomicAdd(&deg[dst[i]], w[i]);
}

__global__ void deg_to_dinv_kernel(float* __restrict__ deg, int n) {
    int i = blockIdx.x * blockDim.x + threadIdx.x;
    if (i < n) {
        float d = deg[i];
        deg[i] = (d > 0.0f) ? rsqrtf(d) : 0.0f;
    }
}

// ---------------------------------------------------------------------------
// FP32 WMMA GEMM: C[M,N] = A[M,64] @ B[64,N].  N is a compile-time constant
// (64 or 32) so every per-k-step stride folds into the instruction's
// immediate offset.  One wave computes one 16x16 C tile via 16 chained
// V_WMMA_F32_16X16X4_F32 ops (K=64).  M % 16 == 0.  wave32; EXEC all-ones.
// ---------------------------------------------------------------------------
template <int N>
__global__ void wmma_gemm_f32_kernel(const float* __restrict__ A,
                                     const float* __restrict__ B,
                                     float* __restrict__ C,
                                     int M) {
    constexpr int tilesN = N >> 4;
    const int lane = threadIdx.x & 31;
    const int wid  = threadIdx.x >> 5;            // wave in block (4 waves)
    const int tile = blockIdx.x * 4 + wid;
    if (tile >= (M >> 4) * tilesN) return;        // uniform per-wave

    const int tm = tile / tilesN;
    const int tn = tile - tm * tilesN;

    const int half   = lane >> 4;                 // 0 | 1
    const int l16    = lane & 15;
    const int ncol   = (tn << 4) + l16;           // B/C column this lane feeds
    const int kShift = half << 1;                 // K offset 0 | 2

    // Per-lane base pointers; all per-step displacements are constants.
    const float* Ap = A + ((tm << 4) + l16) * 64 + kShift;  // +4 floats/step
    const float* Bp = B + kShift * N + ncol;                // +4*N floats/step

    v8f acc = {};
#pragma unroll
    for (int s = 0; s < 16; ++s) {
        v2f a = *(const v2f*)(Ap + s * 4);        // A[m][k0], A[m][k0+1]
        v2f b;
        b.x = Bp[s * 4 * N];                      // B[k0][n]
        b.y = Bp[s * 4 * N + N];                  // B[k0+1][n]
        acc = __builtin_amdgcn_wmma_f32_16x16x4_f32(
            /*neg_a=*/false, a, /*neg_b=*/false, b,
            /*c_mod=*/(short)0, acc, /*reuse_a=*/false, /*reuse_b=*/false);
    }

    // C/D layout: VGPR v, lanes 0-15 -> M=v, lanes 16-31 -> M=v+8; col = l16
    float* Cp = C + ((tm << 4) + (half << 3)) * N + ncol;
#pragma unroll
    for (int v = 0; v < 8; ++v) Cp[v * N] = acc[v];
}

// ---------------------------------------------------------------------------
// Self-loop contribution + bias: out[n,f] = dinv[n]^2 * H[n,f] + bias[f]
// (fully overwrites `out`, so each kernel_launch call is self-initializing)
// ---------------------------------------------------------------------------
template <int F>
__global__ void init_self_kernel(const float* __restrict__ H,
                                 const float* __restrict__ dinv,
                                 const float* __restrict__ bias,
                                 float* __restrict__ out,
                                 int n) {
    int idx = blockIdx.x * blockDim.x + threadIdx.x;
    if (idx >= n * F) return;
    int node = idx >> (F == 64 ? 6 : 5);
    int f    = idx & (F - 1);
    float di = dinv[node];
    out[idx] = di * di * H[idx] + bias[f];
}

// ---------------------------------------------------------------------------
// Edge scatter: one wave per edge; lane f handles feature f (and f+32 when
// F==64).  out[dst] += dinv[src]*w*dinv[dst] * H[src]  via f32 atomics.
// Output rows are L2-resident (<= 12.8 MB vs 192 MB L2).
// ---------------------------------------------------------------------------
template <int F>
__global__ void scatter_edges_kernel(const int* __restrict__ src,
                                     const int* __restrict__ dst,
                                     const float* __restrict__ w,
                                     const float* __restrict__ dinv,
                                     const float* __restrict__ H,
                                     float* __restrict__ out,
                                     int E) {
    int e = blockIdx.x * (blockDim.x >> 5) + (threadIdx.x >> 5);
    if (e >= E) return;                           // uniform per-wave
    int lane = threadIdx.x & 31;
    int s = src[e];
    int d = dst[e];
    float norm = dinv[s] * w[e] * dinv[d];
    const float* hs = H + s * F;                  // 32-bit index math
    float* od = out + d * F;
    atomicAdd(&od[lane], norm * hs[lane]);
    if (F == 64) atomicAdd(&od[lane + 32], norm * hs[lane + 32]);
}

__global__ void relu_kernel(float* __restrict__ x, int n) {
    int i = blockIdx.x * blockDim.x + threadIdx.x;
    if (i < n) x[i] = fmaxf(x[i], 0.0f);
}

// ---------------------------------------------------------------------------
// Launch: x, edge_idx, edge_attr, W1, b1, W2, b2  ->  relu(gcn2(relu(gcn1)))
// ---------------------------------------------------------------------------
extern "C" void kernel_launch(void* const* d_in, const int* in_sizes, int n_in,
                              void* d_out, int out_size, void* d_ws, size_t ws_size,
                              hipStream_t stream) {
    const float* x    = (const float*)d_in[0];
    const int*   eidx = (const int*)  d_in[1];
    const float* ew   = (const float*)d_in[2];
    const float* W1   = (const float*)d_in[3];
    const float* b1   = (const float*)d_in[4];
    const float* W2   = (const float*)d_in[5];
    const float* b2   = (const float*)d_in[6];
    float* out = (float*)d_out;

    const int D_IN = 64, D_HID = 64, D_OUT = 32;
    const int N = in_sizes[0] / D_IN;   // 50000
    const int E = in_sizes[2];          // 1000000
    const int* src = eidx;
    const int* dst = eidx + E;

    // Workspace layout (fp32): dinv[N] | h1[N*64] | a1[N*64] | h2[N*32]
    float* dinv = (float*)d_ws;
    float* h1   = dinv + N;
    float* a1   = h1 + (size_t)N * D_HID;
    float* h2   = a1 + (size_t)N * D_HID;

    const int TPB = 256;
    const int wavesPerBlk = TPB / 32;

    // --- normalization: deg = 1 + sum_in(w); dinv = deg^{-1/2} ---
    deg_init_kernel<<<(N + TPB - 1) / TPB, TPB, 0, stream>>>(dinv, N);
    deg_accum_kernel<<<(E + TPB - 1) / TPB, TPB, 0, stream>>>(dst, ew, dinv, E);
    deg_to_dinv_kernel<<<(N + TPB - 1) / TPB, TPB, 0, stream>>>(dinv, N);

    // --- layer 1: h1 = x @ W1 ; a1 = scatter(norm * h1[src]) + self + b1 ; relu ---
    {
        int tiles = (N / 16) * (D_HID / 16);
        wmma_gemm_f32_kernel<64><<<(tiles + 3) / 4, 128, 0, stream>>>(x, W1, h1, N);
        init_self_kernel<64><<<(N * D_HID + TPB - 1) / TPB, TPB, 0, stream>>>(h1, dinv, b1, a1, N);
        scatter_edges_kernel<64><<<(E + wavesPerBlk - 1) / wavesPerBlk, TPB, 0, stream>>>(
            src, dst, ew, dinv, h1, a1, E);
        relu_kernel<<<(N * D_HID + TPB - 1) / TPB, TPB, 0, stream>>>(a1, N * D_HID);
    }

    // --- layer 2: h2 = a1 @ W2 ; out = scatter(norm * h2[src]) + self + b2 ; relu ---
    {
        int tiles = (N / 16) * (D_OUT / 16);
        wmma_gemm_f32_kernel<32><<<(tiles + 3) / 4, 128, 0, stream>>>(a1, W2, h2, N);
        init_self_kernel<32><<<(N * D_OUT + TPB - 1) / TPB, TPB, 0, stream>>>(h2, dinv, b2, out, N);
        scatter_edges_kernel<32><<<(E + wavesPerBlk - 1) / wavesPerBlk, TPB, 0, stream>>>(
            src, dst, ew, dinv, h2, out, E);
        relu_kernel<<<(N * D_OUT + TPB - 1) / TPB, TPB, 0, stream>>>(out, N * D_OUT);
    }
}